// DeepPolyConvolutionalLayer_48490180772574
// MI455X (gfx1250) — compile-verified
//
#include <hip/hip_runtime.h>

typedef __attribute__((ext_vector_type(2))) float v2f;
typedef __attribute__((ext_vector_type(4))) float v4f;
typedef __attribute__((ext_vector_type(8))) float v8f;

#define C_IN   16
#define HWDIM  16
#define O_OUT  32
#define KTOT   144    // C_IN * 3 * 3
#define NSPAT  256    // 16*16
#define NIN    4096   // C*H*W
#define NOUT   8192   // O*H*W

// ---------------------------------------------------------------------------
// Kernel 1: interval + forward conv as implicit GEMM on V_WMMA_F32_16X16X4_F32
//   out[32x256] = Kmat[32x144] x im2col[144x256]
//   One wave per 16x16 D tile: 2 M-tiles x 16 N-tiles = 32 waves total.
// ---------------------------------------------------------------------------
__global__ __launch_bounds__(256) void dp_conv_wmma(
    const float* __restrict__ x, const float* __restrict__ lb,
    const float* __restrict__ ub, const float* __restrict__ kern,
    const float* __restrict__ bias, float* __restrict__ out)
{
    const int lane = threadIdx.x & 31;
    const int wave = (blockIdx.x * blockDim.x + threadIdx.x) >> 5;   // 0..31
    const int m_base = (wave >> 4) << 4;    // 0 or 16 (output channel tile)
    const int n_base = (wave & 15) << 4;    // spatial tile
    const int half   = lane >> 4;           // 0: lanes 0-15, 1: lanes 16-31
    const int l15    = lane & 15;
    const int m      = m_base + l15;        // A-matrix row (all lanes M=0..15)
    const int n      = n_base + l15;        // B-matrix column
    const int hp     = n >> 4;              // h'
    const int wp     = n & 15;              // w'

    v8f acc_x = {}, acc_l = {}, acc_u = {};

    for (int step = 0; step < KTOT / 4; ++step) {
        const int klo = step * 4 + half * 2;     // this lane holds K=klo, klo+1

        // ---- A fragments: kernel matrix row m, cols klo/klo+1 -------------
        const float a0 = kern[m * KTOT + klo];
        const float a1 = kern[m * KTOT + klo + 1];
        v2f aF = {a0, a1};
        v2f aP = {a0 >= 0.f ? a0 : 0.f, a1 >= 0.f ? a1 : 0.f};
        v2f aN = {a0 <  0.f ? a0 : 0.f, a1 <  0.f ? a1 : 0.f};

        // ---- B fragments: im2col rows klo/klo+1 at column n ---------------
        float bx0 = 0.f, bx1 = 0.f, bl0 = 0.f, bl1 = 0.f, bu0 = 0.f, bu1 = 0.f;
        {
            const int k  = klo;
            const int c  = k / 9, r = k % 9, kh = r / 3, kw = r % 3;
            const int ih = hp + kh - 1, iw = wp + kw - 1;
            if ((unsigned)ih < 16u && (unsigned)iw < 16u) {
                const int idx = c * NSPAT + ih * HWDIM + iw;
                bx0 = x[idx]; bl0 = lb[idx]; bu0 = ub[idx];
            }
        }
        {
            const int k  = klo + 1;
            const int c  = k / 9, r = k % 9, kh = r / 3, kw = r % 3;
            const int ih = hp + kh - 1, iw = wp + kw - 1;
            if ((unsigned)ih < 16u && (unsigned)iw < 16u) {
                const int idx = c * NSPAT + ih * HWDIM + iw;
                bx1 = x[idx]; bl1 = lb[idx]; bu1 = ub[idx];
            }
        }
        v2f bX = {bx0, bx1}, bL = {bl0, bl1}, bU = {bu0, bu1};

        // D = A x B + C   (8 args: neg_a, A, neg_b, B, c_mod, C, reuse_a, reuse_b)
        acc_x = __builtin_amdgcn_wmma_f32_16x16x4_f32(false, aF, false, bX,
                                                      (short)0, acc_x, false, false);
        acc_l = __builtin_amdgcn_wmma_f32_16x16x4_f32(false, aN, false, bU,
                                                      (short)0, acc_l, false, false);
        acc_l = __builtin_amdgcn_wmma_f32_16x16x4_f32(false, aP, false, bL,
                                                      (short)0, acc_l, false, false);
        acc_u = __builtin_amdgcn_wmma_f32_16x16x4_f32(false, aN, false, bL,
                                                      (short)0, acc_u, false, false);
        acc_u = __builtin_amdgcn_wmma_f32_16x16x4_f32(false, aP, false, bU,
                                                      (short)0, acc_u, false, false);
    }

    // ---- epilogue: bias (1x for x_out, 3x for bounds) and store -----------
    float* __restrict__ x_out = out;
    float* __restrict__ l_out = out + NOUT;
    float* __restrict__ u_out = out + 2 * NOUT;
#pragma unroll
    for (int j = 0; j < 8; ++j) {
        const int   mrow = m_base + j + 8 * half;        // D row = output chan
        const float b    = bias[mrow];
        const int   off  = mrow * NSPAT + n_base + l15;
        x_out[off] = acc_x[j] + b;
        l_out[off] = acc_l[j] + 3.f * b;
        u_out[off] = acc_u[j] + 3.f * b;
    }
}

// ---------------------------------------------------------------------------
// Kernel 2: unrolled conv matrix fill [4096 x 8192] — the 128 MB store-bound
// part. One float4 (b128 store) per thread; element is a closed-form kernel
// tap or zero, so single pass, minimal read traffic (~3.5% of threads touch
// the 18 KB kernel, which lives in cache).
// ---------------------------------------------------------------------------
__global__ __launch_bounds__(256) void dp_unroll_weights(
    const float* __restrict__ kern, float* __restrict__ wout)
{
    const int idx  = blockIdx.x * 256 + threadIdx.x;  // float4 index, < 2^23
    const int row  = idx >> 11;                       // n_in  (2048 float4/row)
    const int col4 = idx & 2047;
    const int c = row >> 8, h = (row >> 4) & 15, w = row & 15;
    const int n0 = col4 << 2;                         // n_out of element 0
    const int o = n0 >> 8, hp = (n0 >> 4) & 15, wq = n0 & 15;

    v4f v = {0.f, 0.f, 0.f, 0.f};
    const int dh = h - hp + 1;                        // kh index (pad=1)
    if ((unsigned)dh < 3u) {
        const float* kb = kern + ((o * C_IN + c) * 3 + dh) * 3;
#pragma unroll
        for (int j = 0; j < 4; ++j) {
            const int dw = w - (wq + j) + 1;          // kw index
            if ((unsigned)dw < 3u) v[j] = kb[dw];
        }
    }
    *(v4f*)(wout + (size_t)row * NOUT + n0) = v;
}

// ---------------------------------------------------------------------------
// Kernel 3: bias_vec[o*256 + s] = bias[o]
// ---------------------------------------------------------------------------
__global__ __launch_bounds__(256) void dp_bias_vec(
    const float* __restrict__ bias, float* __restrict__ bout)
{
    const int i = blockIdx.x * 256 + threadIdx.x;     // < 8192
    bout[i] = bias[i >> 8];
}

extern "C" void kernel_launch(void* const* d_in, const int* in_sizes, int n_in,
                              void* d_out, int out_size, void* d_ws, size_t ws_size,
                              hipStream_t stream)
{
    const float* x    = (const float*)d_in[0];   // (1, 4096)
    const float* lb   = (const float*)d_in[1];   // (1, 4096)
    const float* ub   = (const float*)d_in[2];   // (1, 4096)
    // d_in[3] = input_shape (int32, constant [1,16,16,16]) — baked in
    const float* kern = (const float*)d_in[4];   // (32,16,3,3)
    const float* bias = (const float*)d_in[5];   // (32,)

    float* out = (float*)d_out;
    float* wout = out + 3 * NOUT;                       // weights @ 24576
    float* bout = out + 3 * NOUT + (size_t)NIN * NOUT;  // bias_vec tail

    // 32 waves -> 4 blocks x 256 threads; every wave owns one 16x16 D tile.
    dp_conv_wmma<<<4, 256, 0, stream>>>(x, lb, ub, kern, bias, out);

    // 4096*8192/4 float4 stores = 8,388,608 threads -> 32768 blocks of 256.
    dp_unroll_weights<<<32768, 256, 0, stream>>>(kern, wout);

    dp_bias_vec<<<NOUT / 256, 256, 0, stream>>>(bias, bout);
}